// SSCAConv_47631187312880
// MI455X (gfx1250) — compile-verified
//
#include <hip/hip_runtime.h>

typedef float v2f __attribute__((ext_vector_type(2)));
typedef float v8f __attribute__((ext_vector_type(8)));

#define BATCH 8
#define NCH   64
#define HH    128
#define WW    128
#define ROWS_PER_BLOCK 8
#define WAVES 8   // 256 threads / wave32

__launch_bounds__(256)
__global__ void ssca_wmma_kernel(const float* __restrict__ x,
                                 const float* __restrict__ w1,
                                 const float* __restrict__ b1,
                                 const float* __restrict__ w2,
                                 const float* __restrict__ b2,
                                 const float* __restrict__ bias,
                                 float* __restrict__ out)
{
    // input tile with 1-pixel zero halo: rows i0-1..i0+8, cols -1..128
    __shared__ float xs[ROWS_PER_BLOCK + 2][WW + 2];
    // per-wave 16x16 transpose scratch (C-layout -> B-layout), +1 pad vs banks
    __shared__ float trans[WAVES][16][17];

    const int tid  = threadIdx.x;
    const int wv   = tid >> 5;        // wave id 0..7
    const int lane = tid & 31;
    const int half = lane >> 4;       // 0: lanes 0-15, 1: lanes 16-31
    const int n    = lane & 15;       // WMMA column (pixel within tile)

    const int bc = blockIdx.x;
    const int bi = bc / NCH;
    const int c  = bc % NCH;
    const int i0 = blockIdx.y * ROWS_PER_BLOCK;

    // ---- cooperatively stage x tile (zero-padded halo) into LDS ----
    const float* xplane = x + ((size_t)(bi * NCH + c)) * HH * WW;
    for (int idx = tid; idx < (ROWS_PER_BLOCK + 2) * (WW + 2); idx += 256) {
        int r  = idx / (WW + 2);
        int cc = idx % (WW + 2);
        int gr = i0 - 1 + r;
        int gc = cc - 1;
        float val = 0.0f;
        if (gr >= 0 && gr < HH && gc >= 0 && gc < WW)
            val = xplane[gr * WW + gc];
        xs[r][cc] = val;
    }

    // ---- per-channel 9x9 weights into WMMA fp32 A layout (16x4, 3 k-steps) ----
    // A layout: row m = lane%16; vgpr v holds K = 4*s + 2*half + v
    const int m = n;
    v2f a1[3], a2[3];
    for (int s = 0; s < 3; ++s) {
        float a1x = 0.0f, a1y = 0.0f, a2x = 0.0f, a2y = 0.0f;
        int k0 = 4 * s + 2 * half;
        if (m < 9 && k0 < 9)     { a1x = w1[(c * 9 + m) * 9 + k0];
                                   a2x = w2[(c * 9 + m) * 9 + k0]; }
        if (m < 9 && k0 + 1 < 9) { a1y = w1[(c * 9 + m) * 9 + k0 + 1];
                                   a2y = w2[(c * 9 + m) * 9 + k0 + 1]; }
        a1[s].x = a1x; a1[s].y = a1y;
        a2[s].x = a2x; a2[s].y = a2y;
    }
    // fold conv biases into the accumulator init (C layout: row = v + 8*half)
    v8f c1i, c2i;
    for (int v = 0; v < 8; ++v) {
        int row = v + 8 * half;
        c1i[v] = (row < 9) ? b1[c * 9 + row] : 0.0f;
        c2i[v] = (row < 9) ? b2[c * 9 + row] : 0.0f;
    }
    const float bias_c = bias[c];

    __syncthreads();

    const int i = i0 + wv;                       // image row owned by this wave
    float* orow = out + (((size_t)(bi * NCH + c)) * HH + i) * WW;

    for (int jt = 0; jt < WW / 16; ++jt) {
        const int j0 = jt * 16;

        // ---- GEMM 1: depthwise 3x3 == A1[9x9] x patches[9 x 16px] ----
        v8f acc = c1i;
        for (int s = 0; s < 3; ++s) {
            int t0 = 4 * s + 2 * half;           // B layout mirrors A K pattern
            v2f b;
            b.x = (t0     < 9) ? xs[wv + (t0    ) / 3][j0 + n + (t0    ) % 3] : 0.0f;
            b.y = (t0 + 1 < 9) ? xs[wv + (t0 + 1) / 3][j0 + n + (t0 + 1) % 3] : 0.0f;
            acc = __builtin_amdgcn_wmma_f32_16x16x4_f32(false, a1[s], false, b,
                                                        (short)0, acc, false, false);
        }

        // ReLU, then C-layout -> B-layout transpose through per-wave LDS
        for (int v = 0; v < 8; ++v)
            trans[wv][v + 8 * half][n] = fmaxf(acc[v], 0.0f);
        __syncthreads();

        // ---- GEMM 2: grouped 1x1 (9->9) == A2[9x9] x relu(a)[9 x 16px] ----
        v8f acc2 = c2i;
        for (int s = 0; s < 3; ++s) {
            int t0 = 4 * s + 2 * half;           // rows 9..11 are exact zeros
            v2f b;
            b.x = trans[wv][t0    ][n];
            b.y = trans[wv][t0 + 1][n];
            acc2 = __builtin_amdgcn_wmma_f32_16x16x4_f32(false, a2[s], false, b,
                                                         (short)0, acc2, false, false);
        }

        // ---- tanh + unfold-weighted reduction straight from C layout ----
        // lane holds col n, rows v+8*half; rows 9..15 are exactly 0 -> tanh 0
        float partial = 0.0f;
        for (int v = 0; v < 8; ++v) {
            int q = v + 8 * half;
            if (q < 9) {
                float p = xs[wv + q / 3][j0 + n + q % 3];
                partial += tanhf(acc2[v]) * p;
            }
        }
        float total = partial + __shfl_xor(partial, 16, 32);
        if (half == 0)
            orow[j0 + n] = total + bias_c;
        __syncthreads();                         // trans reused next iteration
    }
}

extern "C" void kernel_launch(void* const* d_in, const int* in_sizes, int n_in,
                              void* d_out, int out_size, void* d_ws, size_t ws_size,
                              hipStream_t stream) {
    const float* x    = (const float*)d_in[0];
    const float* w1   = (const float*)d_in[1];
    const float* b1   = (const float*)d_in[2];
    const float* w2   = (const float*)d_in[3];
    const float* b2   = (const float*)d_in[4];
    const float* bias = (const float*)d_in[5];
    float* out = (float*)d_out;

    dim3 grid(BATCH * NCH, HH / ROWS_PER_BLOCK);
    ssca_wmma_kernel<<<grid, 256, 0, stream>>>(x, w1, b1, w2, b2, bias, out);
}